// ValuesAndGraphStructure_68247030333768
// MI455X (gfx1250) — compile-verified
//
#include <hip/hip_runtime.h>
#include <math.h>

// Problem dims (from reference): B=64, N=1024, P=32, L1=512, L2=128, C=1
#define B_  64
#define N_  1024
#define P_  32
#define L1_ 512
#define L2_ 128
#define K1_ (N_ * P_)      // 32768 = fc1 reduction dim
#define KSPLIT_ 16         // fc1 K-chunks
#define KCH_ (K1_ / KSPLIT_) // 2048 per chunk

typedef __attribute__((ext_vector_type(2))) float v2f;
typedef __attribute__((ext_vector_type(4))) float v4f;
typedef __attribute__((ext_vector_type(8))) float v8f;

__device__ __forceinline__ float wave_reduce_add(float p) {
  // butterfly: every lane ends with the full sum (wave32)
  for (int off = 16; off > 0; off >>= 1) p += __shfl_xor(p, off, 32);
  return p;
}

// ---------------------------------------------------------------------------
// Pass 1 over A (256 MB, non-temporal): row sums -> dinv = deg^-1/2
// (deg==0 -> 1), fused y = dinv*x.  One wave per row, 8 rows per block.
// No LDS, no barriers: shfl butterfly reduction.
// ---------------------------------------------------------------------------
__global__ void k_deg(const float* __restrict__ A, const float* __restrict__ x,
                      float* __restrict__ dinv, float* __restrict__ y) {
  const int wave = threadIdx.x >> 5;
  const int lane = threadIdx.x & 31;
  const int row  = blockIdx.x * 8 + wave;     // 0 .. B*N-1
  const v4f* __restrict__ base = (const v4f*)(A + (size_t)row * N_);
  float s = 0.0f;
#pragma unroll
  for (int i = 0; i < 8; ++i) {               // 32 contiguous v4f per wave pass
    v4f a4 = __builtin_nontemporal_load(base + i * 32 + lane);
    s += a4.x + a4.y + a4.z + a4.w;
  }
  s = wave_reduce_add(s);
  if (lane == 0) {
    float d  = (s != 0.0f) ? s : 1.0f;
    float di = rsqrtf(d);
    dinv[row] = di;
    y[row]    = di * x[row];
  }
}

// ---------------------------------------------------------------------------
// Pass 2 over A (non-temporal): h = dinv[i]*(A[i,:].y[b,:]) + alpha*x[i],
// fused Linear(1,P)+ReLU: hpre[row*P+p] = relu(h*Wpre[p]+bpre[p]).
// y[b,:] (4 KB) staged in LDS once per block; one wave per row, 8 rows/block
// (all 8 rows share the same batch b since 1024 % 8 == 0).
// Butterfly reduce leaves h in all 32 lanes == P, so the expansion needs no
// barrier or broadcast.
// ---------------------------------------------------------------------------
__global__ void k_matvec_pre(const float* __restrict__ A, const float* __restrict__ x,
                             const float* __restrict__ y, const float* __restrict__ dinv,
                             const float* __restrict__ alpha,
                             const float* __restrict__ Wpre, const float* __restrict__ bpre,
                             float* __restrict__ hpre) {
  const int wave = threadIdx.x >> 5;
  const int lane = threadIdx.x & 31;
  const int row  = blockIdx.x * 8 + wave;     // b*N + i
  const int b    = blockIdx.x >> 7;           // 128 blocks per batch
  __shared__ float ylds[N_];
  ((v4f*)ylds)[threadIdx.x] = ((const v4f*)(y + (size_t)b * N_))[threadIdx.x];
  __syncthreads();
  const v4f* __restrict__ base = (const v4f*)(A + (size_t)row * N_);
  const v4f* __restrict__ yb   = (const v4f*)ylds;
  float s = 0.0f;
#pragma unroll
  for (int i = 0; i < 8; ++i) {
    v4f a4 = __builtin_nontemporal_load(base + i * 32 + lane);
    v4f y4 = yb[i * 32 + lane];
    s += a4.x * y4.x + a4.y * y4.y + a4.z * y4.z + a4.w * y4.w;
  }
  s = wave_reduce_add(s);
  const float hval = dinv[row] * s + alpha[0] * x[row];   // all lanes
  const float v = hval * Wpre[lane] + bpre[lane];         // lane == p (P==32)
  hpre[(size_t)row * P_ + lane] = fmaxf(v, 0.0f);
}

// ---------------------------------------------------------------------------
// fc1 partial GEMM via V_WMMA_F32_16X16X4_F32, K split into KSPLIT chunks for
// parallelism (2048 waves instead of 128 -> enough outstanding loads to cover
// the 67 MB W1 stream).  One wave per (16x16 tile, K-chunk); dual accumulators
// break the WMMA D->C chain.  Partials (no bias/relu) land in workspace.
// A frag (16x4 f32): lane m = lane&15, contiguous float2 at koff=(lane>>4)*2.
// B frag (4x16): mirrored layout over W1 rows (B = W1^T).
// C/D (16x16): VGPR r -> M = r + 8*(lane>>4), Ncol = lane&15.
// ---------------------------------------------------------------------------
__global__ void k_fc1_wmma(const float* __restrict__ Aact, const float* __restrict__ W1,
                           float* __restrict__ partial) {
  const int lane  = threadIdx.x;             // 0..31 (wave32)
  const int n0    = blockIdx.x * 16;         // L1 tile
  const int m0    = blockIdx.y * 16;         // batch tile
  const int kbase = blockIdx.z * KCH_;       // K chunk
  const int r     = lane & 15;
  const int koff  = (lane >> 4) << 1;        // 0 or 2
  const float* ap = Aact + (size_t)(m0 + r) * K1_ + kbase + koff;
  const float* bp = W1   + (size_t)(n0 + r) * K1_ + kbase + koff;
  v8f acc0 = {};
  v8f acc1 = {};
#pragma unroll 4
  for (int k = 0; k < KCH_; k += 8) {
    v2f a0 = *(const v2f*)(ap + k);
    v2f b0 = *(const v2f*)(bp + k);
    acc0 = __builtin_amdgcn_wmma_f32_16x16x4_f32(
        /*neg_a=*/false, a0, /*neg_b=*/false, b0,
        /*c_mod=*/(short)0, acc0, /*reuse_a=*/false, /*reuse_b=*/false);
    v2f a1v = *(const v2f*)(ap + k + 4);
    v2f b1v = *(const v2f*)(bp + k + 4);
    acc1 = __builtin_amdgcn_wmma_f32_16x16x4_f32(
        false, a1v, false, b1v, (short)0, acc1, false, false);
  }
  const int ncol  = n0 + (lane & 15);
  const int mbase = m0 + ((lane >> 4) << 3);
  float* dst = partial + (size_t)blockIdx.z * (B_ * L1_);
#pragma unroll
  for (int rr = 0; rr < 8; ++rr)
    dst[(size_t)(mbase + rr) * L1_ + ncol] = acc0[rr] + acc1[rr];
}

// ---------------------------------------------------------------------------
// fc1 reduce across K-chunks + bias + relu -> a1[64,512]
// ---------------------------------------------------------------------------
__global__ void k_fc1_reduce(const float* __restrict__ partial, const float* __restrict__ b1,
                             float* __restrict__ a1) {
  const int tid = blockIdx.x * blockDim.x + threadIdx.x;  // 0 .. B*L1-1
  const int l = tid & (L1_ - 1);
  float s = b1[l];
#pragma unroll
  for (int c = 0; c < KSPLIT_; ++c) s += partial[(size_t)c * (B_ * L1_) + tid];
  a1[tid] = fmaxf(s, 0.0f);
}

// ---------------------------------------------------------------------------
// fc2: a2[64,128] = relu(a1 @ W2^T + b2). 4M MACs -> plain VALU.
// ---------------------------------------------------------------------------
__global__ void k_fc2(const float* __restrict__ a1, const float* __restrict__ W2,
                      const float* __restrict__ b2, float* __restrict__ a2) {
  const int tid = blockIdx.x * blockDim.x + threadIdx.x;  // 0 .. B*L2-1
  const int m = tid >> 7;        // / L2_
  const int l = tid & (L2_ - 1);
  const v4f* av = (const v4f*)(a1 + (size_t)m * L1_);
  const v4f* wv = (const v4f*)(W2 + (size_t)l * L1_);
  float acc = 0.0f;
#pragma unroll 4
  for (int k = 0; k < L1_ / 4; ++k) {
    v4f a = av[k], w = wv[k];
    acc += a.x * w.x + a.y * w.y + a.z * w.z + a.w * w.w;
  }
  a2[tid] = fmaxf(acc + b2[l], 0.0f);
}

// ---------------------------------------------------------------------------
// fc3 + sigmoid: out[64] = sigmoid(a2 @ W3^T + b3). Tiny.
// ---------------------------------------------------------------------------
__global__ void k_fc3(const float* __restrict__ a2, const float* __restrict__ W3,
                      const float* __restrict__ b3, float* __restrict__ out) {
  const int m = threadIdx.x;
  if (m >= B_) return;
  float acc = 0.0f;
#pragma unroll 4
  for (int k = 0; k < L2_; ++k) acc += a2[m * L2_ + k] * W3[k];
  float z = acc + b3[0];
  out[m] = 1.0f / (1.0f + __expf(-z));
}

extern "C" void kernel_launch(void* const* d_in, const int* in_sizes, int n_in,
                              void* d_out, int out_size, void* d_ws, size_t ws_size,
                              hipStream_t stream) {
  const float* x     = (const float*)d_in[0];   // [B, N]
  const float* A     = (const float*)d_in[1];   // [B, N, N]
  const float* alpha = (const float*)d_in[2];   // [1]
  const float* Wpre  = (const float*)d_in[3];   // [P, 1]
  const float* bpre  = (const float*)d_in[4];   // [P]
  const float* W1    = (const float*)d_in[5];   // [L1, N*P]
  const float* b1    = (const float*)d_in[6];   // [L1]
  const float* W2    = (const float*)d_in[7];   // [L2, L1]
  const float* b2    = (const float*)d_in[8];   // [L2]
  const float* W3    = (const float*)d_in[9];   // [C, L2]
  const float* b3    = (const float*)d_in[10];  // [C]
  float* out = (float*)d_out;                   // [B, C] = 64 floats

  // Workspace layout (floats): ~11.2 MB total
  float* ws      = (float*)d_ws;
  float* dinv    = ws;                            // B*N          = 65536
  float* y       = dinv    + (size_t)B_ * N_;     // B*N          = 65536
  float* hpre    = y       + (size_t)B_ * N_;     // B*N*P        = 2097152
  float* partial = hpre    + (size_t)B_ * N_ * P_;// KSPLIT*B*L1  = 524288
  float* a1      = partial + (size_t)KSPLIT_ * B_ * L1_; // B*L1  = 32768
  float* a2      = a1      + (size_t)B_ * L1_;    // B*L2         = 8192

  // Pass 1 over A (NT): degrees -> dinv, y = dinv * x.  8 waves/block, 1 row/wave.
  k_deg<<<(B_ * N_) / 8, 256, 0, stream>>>(A, x, dinv, y);

  // Pass 2 over A (NT): normalized matvec + alpha*x, fused Linear(1,P)+ReLU
  k_matvec_pre<<<(B_ * N_) / 8, 256, 0, stream>>>(A, x, y, dinv, alpha, Wpre, bpre, hpre);

  // fc1 partials via f32 WMMA: (L1/16, B/16, KSPLIT) single-wave workgroups
  dim3 g3(L1_ / 16, B_ / 16, KSPLIT_);
  k_fc1_wmma<<<g3, 32, 0, stream>>>(hpre, W1, partial);

  // fc1 reduce + bias + relu
  k_fc1_reduce<<<(B_ * L1_) / 256, 256, 0, stream>>>(partial, b1, a1);

  // fc2 + relu
  k_fc2<<<(B_ * L2_) / 256, 256, 0, stream>>>(a1, W2, b2, a2);

  // fc3 + sigmoid
  k_fc3<<<1, 64, 0, stream>>>(a2, W3, b3, out);
}